// CapsuleOut_64647847739773
// MI455X (gfx1250) — compile-verified
//
#include <hip/hip_runtime.h>
#include <hip/hip_bf16.h>
#include <math.h>

#define B_TOT 256
#define O_DIM 10
#define R_DIM 1152
#define I_DIM 8
#define D_DIM 16
#define NITER 3

typedef float v2f __attribute__((ext_vector_type(2)));
typedef float v8f __attribute__((ext_vector_type(8)));
typedef int   i32x4 __attribute__((ext_vector_type(4)));
typedef int   i32x8 __attribute__((ext_vector_type(8)));
typedef unsigned int u32x4 __attribute__((ext_vector_type(4)));

// ---- CDNA5 data-mover feature detection (device pass only) -----------------
#if defined(__HIP_DEVICE_COMPILE__)
  #if __has_builtin(__builtin_amdgcn_tensor_load_to_lds)
    #if __has_include(<hip/amd_detail/amd_gfx1250_TDM.h>)
      #define TDM_ARITY 6          /* amdgpu-toolchain: 6-arg builtin */
    #else
      #define TDM_ARITY 5          /* ROCm 7.2: 5-arg builtin */
    #endif
  #else
    #define TDM_ARITY 0
  #endif
  #if (TDM_ARITY == 0) && \
      __has_builtin(__builtin_amdgcn_global_load_async_to_lds_b128) && \
      __has_builtin(__builtin_amdgcn_s_wait_asynccnt)
    #define USE_ASYNC_LDS 1
  #else
    #define USE_ASYNC_LDS 0
  #endif
#else
  #define TDM_ARITY 0
  #define USE_ASYNC_LDS 0
#endif

// ---------------------------------------------------------------------------
// Kernel 1: priors via V_WMMA_F32_16X16X4_F32.
// For fixed (o, r):  priors[b, d] = sum_i in[b, r, i] * W[o, r, i, d]
// is a [256 x 8] @ [8 x 16] GEMM. One workgroup (8 waves) per (o, r); each
// wave computes 16x16 output tiles over b with two K=4 WMMAs (K total = 8).
// ---------------------------------------------------------------------------
__global__ __launch_bounds__(256) void capsule_priors_wmma(
    const float* __restrict__ in,   // [B, R, I]
    const float* __restrict__ W,    // [O, R, I, D]
    float* __restrict__ P,          // [n_b, O, R, D] chunk
    int b_base, int n_b)
{
  const int gid  = blockIdx.x;
  const int r    = gid % R_DIM;
  const int o    = gid / R_DIM;
  const int lane = threadIdx.x & 31;
  const int wave = threadIdx.x >> 5;
  const int half = lane >> 4;      // 0: lanes 0-15, 1: lanes 16-31
  const int l    = lane & 15;
  const int k0   = half * 2;       // K pair owned by this half-wave

  // B operands (shared across every b tile of this (o, r)).
  const float* wb = W + ((size_t)o * R_DIM + r) * (I_DIM * D_DIM);
  v2f b_lo, b_hi;
  b_lo.x = wb[(k0 + 0) * D_DIM + l];
  b_lo.y = wb[(k0 + 1) * D_DIM + l];
  b_hi.x = wb[(k0 + 4) * D_DIM + l];
  b_hi.y = wb[(k0 + 5) * D_DIM + l];

  const int ntiles = n_b >> 4;
  for (int t = wave; t < ntiles; t += 8) {
    const int b0 = t * 16;
    const float* ab = in + ((size_t)(b_base + b0 + l) * R_DIM + r) * I_DIM;
    v2f a_lo, a_hi;
    a_lo.x = ab[k0 + 0]; a_lo.y = ab[k0 + 1];
    a_hi.x = ab[k0 + 4]; a_hi.y = ab[k0 + 5];

    v8f acc = {};
    acc = __builtin_amdgcn_wmma_f32_16x16x4_f32(false, a_lo, false, b_lo,
                                                (short)0, acc, false, false);
    acc = __builtin_amdgcn_wmma_f32_16x16x4_f32(false, a_hi, false, b_hi,
                                                (short)0, acc, false, false);
#pragma unroll
    for (int v = 0; v < 8; ++v) {
      const int m = v + half * 8;
      P[(((size_t)(b0 + m) * O_DIM + o) * R_DIM + r) * D_DIM + l] = acc[v];
    }
  }
}

// ---------------------------------------------------------------------------
// Workgroup reductions (wave32 shuffles + tiny LDS combine).
// ---------------------------------------------------------------------------
__device__ __forceinline__ float block_max(float v, volatile float* sRed) {
#pragma unroll
  for (int off = 16; off >= 1; off >>= 1)
    v = fmaxf(v, __shfl_xor(v, off, 32));
  if ((threadIdx.x & 31) == 0) sRed[threadIdx.x >> 5] = v;
  __syncthreads();
  float r = sRed[0];
#pragma unroll
  for (int i = 1; i < 8; ++i) r = fmaxf(r, sRed[i]);
  __syncthreads();
  return r;
}

__device__ __forceinline__ float block_sum(float v, volatile float* sRed) {
#pragma unroll
  for (int off = 16; off >= 1; off >>= 1)
    v += __shfl_xor(v, off, 32);
  if ((threadIdx.x & 31) == 0) sRed[threadIdx.x >> 5] = v;
  __syncthreads();
  float r = 0.f;
#pragma unroll
  for (int i = 0; i < 8; ++i) r += sRed[i];
  __syncthreads();
  return r;
}

// ---------------------------------------------------------------------------
// Dynamic routing over an LDS-resident P slice [R_DIM x D_DIM] for one (b,o).
// Logits are provably constant along D, so state is L[r] only.
// ---------------------------------------------------------------------------
__device__ __forceinline__ void capsule_route(float* sP, float* out_g)
{
  __shared__ float sL[R_DIM];
  __shared__ float sE[R_DIM];
  __shared__ float sRed[8];
  __shared__ float sT8[8 * D_DIM];
  __shared__ float sT[D_DIM];

  const int tid = threadIdx.x;
  for (int r = tid; r < R_DIM; r += 256) sL[r] = 0.f;
  __syncthreads();

  float scale = 0.f;
  float outv[D_DIM];

  for (int it = 0; it < NITER; ++it) {
    // softmax over r
    float m = -3.0e38f;
    for (int r = tid; r < R_DIM; r += 256) m = fmaxf(m, sL[r]);
    m = block_max(m, sRed);

    float ps = 0.f;
    for (int r = tid; r < R_DIM; r += 256) {
      float e = expf(sL[r] - m);
      sE[r] = e;
      ps += e;
    }
    const float S = block_sum(ps, sRed);   // its barriers also publish sE
    const float invS = 1.0f / S;

    // t[d] = sum_r e[r] * P[r, d]; thread owns d = tid&15, rows (tid>>4)+16k
    {
      const int d = tid & 15;
      float acc = 0.f;
      for (int r = (tid >> 4); r < R_DIM; r += 16)
        acc += sE[r] * sP[r * D_DIM + d];
      acc += __shfl_xor(acc, 16, 32);
      if ((tid & 31) < 16) sT8[(tid >> 5) * D_DIM + d] = acc;
      __syncthreads();
      if (tid < D_DIM) {
        float td = 0.f;
#pragma unroll
        for (int w = 0; w < 8; ++w) td += sT8[w * D_DIM + tid];
        sT[tid] = td * invS;               // s[d] (pre-squash)
      }
      __syncthreads();
    }

    // squash, computed redundantly by every thread
    float sq = 0.f;
#pragma unroll
    for (int d = 0; d < D_DIM; ++d) { float sd = sT[d]; outv[d] = sd; sq += sd * sd; }
    scale = (sq / (1.0f + sq)) / sqrtf(sq);
#pragma unroll
    for (int d = 0; d < D_DIM; ++d) outv[d] *= scale;

    // logits update (skip after the final outputs are formed)
    if (it < NITER - 1) {
      for (int r = tid; r < R_DIM; r += 256) {
        float delta = 0.f;
#pragma unroll
        for (int d = 0; d < D_DIM; ++d) delta = fmaf(sP[r * D_DIM + d], outv[d], delta);
        sL[r] += delta;
      }
    }
  }
  if (tid < D_DIM) out_g[tid] = sT[tid] * scale;
}

// ---------------------------------------------------------------------------
// Kernel 2: pull the 72 KB P slice for one (b,o) into LDS, run routing.
// Preferred fill path: one TDM tensor_load_to_lds descriptor per workgroup
// (issued by wave 0, drained with s_wait_tensorcnt). Fallbacks: per-lane
// async-to-LDS b128 copies (ASYNCcnt engine), then plain vector loads.
// ---------------------------------------------------------------------------
__global__ __launch_bounds__(256) void capsule_routing(
    const float* __restrict__ P, float* __restrict__ out, int b_base)
{
  __shared__ float sP[R_DIM * D_DIM];
  const int bo = blockIdx.x;          // b_local * O + o
  const float* src = P + (size_t)bo * (R_DIM * D_DIM);

#if TDM_ARITY > 0
  if (threadIdx.x < 32) {             // one wave issues the DMA (per-wave op)
    const unsigned NELEM = R_DIM * D_DIM;                  // 18432 f32
    unsigned long long ga = (unsigned long long)(const void*)src;
    unsigned lds_off =
        (unsigned)(unsigned long long)(__attribute__((address_space(3))) char*)sP;

    u32x4 g0;
    g0.x = 1u;                                             // count=1, user D#
    g0.y = lds_off;                                        // lds_addr
    g0.z = (unsigned)(ga & 0xffffffffu);                   // global_addr[31:0]
    g0.w = (unsigned)((ga >> 32) & 0x01ffffffu)            // global_addr[56:32]
         | (2u << 30);                                     // type = 2 (image)

    i32x8 g1;
    g1[0] = (int)(2u << 16);                 // data_size = 4B, no mask/pad
    g1[1] = (int)((NELEM & 0xffffu) << 16);  // tensor_dim0[15:0]  (bits 63:48)
    g1[2] = (int)((NELEM >> 16) | (1u << 16)); // dim0[31:16] | tensor_dim1=1
    g1[3] = (int)((NELEM & 0xffffu) << 16);  // tile_dim0 = 18432 (bits 127:112)
    g1[4] = 1;                               // tile_dim1 = 1, tile_dim2 = 0
    g1[5] = (int)NELEM;                      // tensor_dim0_stride[31:0]
    g1[6] = 0;                               // stride upper bits
    g1[7] = 0;
    i32x4 g2 = {0, 0, 0, 0};
    i32x4 g3 = {0, 0, 0, 0};
#if TDM_ARITY == 6
    i32x8 g4 = {0, 0, 0, 0, 0, 0, 0, 0};
    __builtin_amdgcn_tensor_load_to_lds(g0, g1, g2, g3, g4, 0);
#else
    __builtin_amdgcn_tensor_load_to_lds(g0, g1, g2, g3, 0);
#endif
#if __has_builtin(__builtin_amdgcn_s_wait_tensorcnt)
    __builtin_amdgcn_s_wait_tensorcnt(0);
#else
    asm volatile("s_wait_tensorcnt 0x0" ::: "memory");
#endif
  }
  __syncthreads();
#elif USE_ASYNC_LDS
  {
    const int n128 = (R_DIM * D_DIM) / 4;  // 4608 x 16B
    __attribute__((address_space(1))) char* gp =
        (__attribute__((address_space(1))) char*)(float*)src;
    __attribute__((address_space(3))) char* lp =
        (__attribute__((address_space(3))) char*)sP;
    for (int i = threadIdx.x; i < n128; i += 256) {
      __builtin_amdgcn_global_load_async_to_lds_b128(
          (__attribute__((address_space(1))) i32x4*)(gp + (size_t)i * 16),
          (__attribute__((address_space(3))) i32x4*)(lp + i * 16), 0, 0);
    }
    __builtin_amdgcn_s_wait_asynccnt(0);
  }
  __syncthreads();
#else
  {
    const float4* s4 = (const float4*)src;
    float4* d4 = (float4*)sP;
    for (int i = threadIdx.x; i < (R_DIM * D_DIM) / 4; i += 256) d4[i] = s4[i];
  }
  __syncthreads();
#endif

  capsule_route(sP, out + ((size_t)b_base * O_DIM + bo) * D_DIM);
}

// ---------------------------------------------------------------------------
// Fallback: fully fused (scalar priors into LDS) if workspace is too small.
// ---------------------------------------------------------------------------
__global__ __launch_bounds__(256) void capsule_fused(
    const float* __restrict__ in, const float* __restrict__ W,
    float* __restrict__ out)
{
  __shared__ float sP[R_DIM * D_DIM];
  const int b = blockIdx.x / O_DIM;
  const int o = blockIdx.x % O_DIM;
  for (int rd = threadIdx.x; rd < R_DIM * D_DIM; rd += 256) {
    const int r = rd >> 4, d = rd & 15;
    const float* xi = in + ((size_t)b * R_DIM + r) * I_DIM;
    const float* wi = W + ((size_t)o * R_DIM + r) * (I_DIM * D_DIM) + d;
    float acc = 0.f;
#pragma unroll
    for (int i = 0; i < I_DIM; ++i) acc = fmaf(xi[i], wi[i * D_DIM], acc);
    sP[rd] = acc;
  }
  __syncthreads();
  capsule_route(sP, out + (size_t)blockIdx.x * D_DIM);
}

// ---------------------------------------------------------------------------
extern "C" void kernel_launch(void* const* d_in, const int* in_sizes, int n_in,
                              void* d_out, int out_size, void* d_ws, size_t ws_size,
                              hipStream_t stream)
{
  (void)in_sizes; (void)n_in; (void)out_size;
  const float* in = (const float*)d_in[0];
  const float* W  = (const float*)d_in[1];
  float* out = (float*)d_out;

  // Largest power-of-two b-chunk whose priors slab fits the workspace.
  int bchunk = 0;
  for (int c = B_TOT; c >= 16; c >>= 1) {
    size_t need = (size_t)c * O_DIM * R_DIM * D_DIM * sizeof(float);
    if (need <= ws_size) { bchunk = c; break; }
  }

  if (bchunk > 0) {
    float* P = (float*)d_ws;
    for (int b0 = 0; b0 < B_TOT; b0 += bchunk) {
      const int nb = (B_TOT - b0 < bchunk) ? (B_TOT - b0) : bchunk;
      capsule_priors_wmma<<<dim3(O_DIM * R_DIM), dim3(256), 0, stream>>>(
          in, W, P, b0, nb);
      capsule_routing<<<dim3(nb * O_DIM), dim3(256), 0, stream>>>(P, out, b0);
    }
  } else {
    capsule_fused<<<dim3(B_TOT * O_DIM), dim3(256), 0, stream>>>(in, W, out);
  }
}